// GIN_61349312856765
// MI455X (gfx1250) — compile-verified
//
#include <hip/hip_runtime.h>

#define N_NODES   100000
#define N_EDGES   1600000
#define N_GRAPHS  1000
#define D_FEAT    128
#define HID       64
#define BN_EPS    1e-5f
#define NTILES    (N_NODES / 16)   // 6250, exact

typedef __attribute__((ext_vector_type(2))) float v2f;
typedef __attribute__((ext_vector_type(8))) float v8f;

// ---------------------------------------------------------------------------
// Fused 2-layer MLP: hout = ReLU(BN2(ReLU(BN1(x@W1+b1)) @ W2 + b2))
// One wave computes a 16-node tile using V_WMMA_F32_16X16X4_F32.
// Weights are cooperatively pre-swizzled into LDS in exact B-fragment order.
// ---------------------------------------------------------------------------
template<int K1>
__launch_bounds__(128)
__global__ void gin_mlp_kernel(const float* __restrict__ xin,   // [N, K1]
                               const float* __restrict__ W1,    // [K1, 64]
                               const float* __restrict__ b1,    // [64]
                               const float* __restrict__ bn1,   // [4, 64]
                               const float* __restrict__ W2,    // [64, 64]
                               const float* __restrict__ b2,    // [64]
                               const float* __restrict__ bn2,   // [4, 64]
                               float* __restrict__ hout,        // [N, 64]
                               int ntiles) {
  constexpr int KS1 = K1 / 4;
  __shared__ __align__(16) float w1s[4 * KS1 * 32 * 2];
  __shared__ __align__(16) float w2s[4 * 16 * 32 * 2];
  __shared__ __align__(16) float tb[4][16 * 68];   // padded (68) to kill bank conflicts

  const int tid = threadIdx.x;

  // swizzle W1 into B-fragment layout: [ntile][kstep][lane]{2 comps}
  for (int i = tid; i < K1 * 64; i += 128) {
    int kk = i >> 6, nn = i & 63;
    int idx = ((((nn >> 4) * KS1) + (kk >> 2)) * 32 + (nn & 15) + ((kk & 2) << 3)) * 2 + (kk & 1);
    w1s[idx] = W1[i];
  }
  for (int i = tid; i < 64 * 64; i += 128) {
    int kk = i >> 6, nn = i & 63;
    int idx = ((((nn >> 4) * 16) + (kk >> 2)) * 32 + (nn & 15) + ((kk & 2) << 3)) * 2 + (kk & 1);
    w2s[idx] = W2[i];
  }
  __syncthreads();

  const int lane = tid & 31;
  const int wave = tid >> 5;
  const int row  = lane & 15;
  const int hi   = lane >> 4;
  const int tile = blockIdx.x * 4 + wave;
  const bool active = tile < ntiles;
  float* tbuf = tb[wave];

  const v8f vzero = {0.f, 0.f, 0.f, 0.f, 0.f, 0.f, 0.f, 0.f};

  // ---- GEMM1 (K1-deep) + bias + BN1 + ReLU -> tbuf ----
  if (active) {
    v2f a1[KS1];
    const float* xr = xin + (size_t)(tile * 16 + row) * K1 + 2 * hi;
#pragma unroll
    for (int k = 0; k < KS1; k++) a1[k] = *(const v2f*)(xr + 4 * k);

#pragma unroll
    for (int nt = 0; nt < 4; nt++) {
      v8f acc = vzero;
      const v2f* wf = ((const v2f*)w1s) + nt * KS1 * 32 + lane;
#pragma unroll
      for (int k = 0; k < KS1; k++) {
        acc = __builtin_amdgcn_wmma_f32_16x16x4_f32(false, a1[k], false, wf[k * 32],
                                                    (short)0, acc, false, false);
      }
      int n = nt * 16 + row;                    // output column
      float s  = bn1[n] * rsqrtf(bn1[192 + n] + BN_EPS);
      float t0 = bn1[64 + n] + (b1[n] - bn1[128 + n]) * s;
#pragma unroll
      for (int i = 0; i < 8; i++) {
        float y = acc[i] * s + t0;
        tbuf[(i + 8 * hi) * 68 + n] = fmaxf(y, 0.0f);
      }
    }
  }
  __syncthreads();

  // ---- GEMM2 (64-deep) from tbuf ----
  v8f acc2[4];
  if (active) {
#pragma unroll
    for (int nt = 0; nt < 4; nt++) {
      v8f acc = vzero;
      const v2f* wf = ((const v2f*)w2s) + nt * 16 * 32 + lane;
      const float* tr = tbuf + row * 68 + 2 * hi;
#pragma unroll
      for (int k = 0; k < 16; k++) {
        v2f a = *(const v2f*)(tr + 4 * k);
        acc = __builtin_amdgcn_wmma_f32_16x16x4_f32(false, a, false, wf[k * 32],
                                                    (short)0, acc, false, false);
      }
      acc2[nt] = acc;
    }
  }
  __syncthreads();

  // ---- bias + BN2 + ReLU -> tbuf (for coalesced store) ----
  if (active) {
#pragma unroll
    for (int nt = 0; nt < 4; nt++) {
      int n = nt * 16 + row;
      float s  = bn2[n] * rsqrtf(bn2[192 + n] + BN_EPS);
      float t0 = bn2[64 + n] + (b2[n] - bn2[128 + n]) * s;
#pragma unroll
      for (int i = 0; i < 8; i++) {
        float y = acc2[nt][i] * s + t0;
        tbuf[(i + 8 * hi) * 68 + n] = fmaxf(y, 0.0f);
      }
    }
  }
  __syncthreads();

  // ---- coalesced 256B row stores ----
  if (active) {
#pragma unroll
    for (int j = 0; j < 8; j++) {
      int flat = j * 32 + lane;
      int r = flat >> 4;
      int c = (flat & 15) << 2;
      float4 v = *(const float4*)(tbuf + r * 68 + c);
      *(float4*)(hout + (size_t)(tile * 16 + r) * 64 + c) = v;
    }
  }
}

// ---------------------------------------------------------------------------
// agg[dst] += h[src]   (edge-parallel, float4 gather + 4 f32 atomics)
// ---------------------------------------------------------------------------
__global__ void gin_scatter_kernel(const float* __restrict__ h,
                                   const int* __restrict__ src,
                                   const int* __restrict__ dst,
                                   float* __restrict__ agg) {
  long long idx = (long long)blockIdx.x * blockDim.x + threadIdx.x;
  int e = (int)(idx >> 4);
  if (e >= N_EDGES) return;
  int c = ((int)idx & 15) << 2;
  int s = src[e], d = dst[e];
  const float4 v = *(const float4*)(h + (size_t)s * 64 + c);
  float* p = agg + (size_t)d * 64 + c;
  atomicAdd(p + 0, v.x);
  atomicAdd(p + 1, v.y);
  atomicAdd(p + 2, v.z);
  atomicAdd(p + 3, v.w);
}

// ---------------------------------------------------------------------------
// pooled[batch[n]] += h[n]; optionally counts[batch[n]] += 1
// ---------------------------------------------------------------------------
__global__ void gin_pool_kernel(const float* __restrict__ h,
                                const int* __restrict__ batch,
                                float* __restrict__ pooled,
                                float* __restrict__ counts,
                                int addCount) {
  long long idx = (long long)blockIdx.x * blockDim.x + threadIdx.x;
  int n = (int)(idx >> 4);
  if (n >= N_NODES) return;
  int c = ((int)idx & 15) << 2;
  int g = batch[n];
  const float4 v = *(const float4*)(h + (size_t)n * 64 + c);
  float* p = pooled + (size_t)g * 64 + c;
  atomicAdd(p + 0, v.x);
  atomicAdd(p + 1, v.y);
  atomicAdd(p + 2, v.z);
  atomicAdd(p + 3, v.w);
  if (addCount && (((int)idx & 15) == 0)) atomicAdd(counts + g, 1.0f);
}

// ---------------------------------------------------------------------------
// out[g][t] = sum_l pooled[l][g] @ lin_W[l][:,t]  + counts[g]*lin_b[0][t]
//             + sum_{l>=1} lin_b[l][t]
// ---------------------------------------------------------------------------
__global__ void gin_final_kernel(const float* __restrict__ pooled,  // [5,1000,64]
                                 const float* __restrict__ counts,  // [1000]
                                 const float* __restrict__ linW,    // [5,64,10]
                                 const float* __restrict__ linb,    // [5,10]
                                 float* __restrict__ out) {         // [1000,10]
  int idx = blockIdx.x * blockDim.x + threadIdx.x;
  if (idx >= N_GRAPHS * 10) return;
  int g = idx / 10, t = idx - g * 10;
  float acc = counts[g] * linb[t];
  for (int l = 1; l < 5; l++) acc += linb[l * 10 + t];
  for (int l = 0; l < 5; l++) {
    const float* p = pooled + ((size_t)l * N_GRAPHS + g) * 64;
    const float* w = linW + l * 640 + t;
    float s = 0.f;
#pragma unroll
    for (int k = 0; k < 64; k++) s += p[k] * w[k * 10];
    acc += s;
  }
  out[idx] = acc;
}

__global__ void gin_zero_kernel(float* __restrict__ p, int n) {
  int i = blockIdx.x * blockDim.x + threadIdx.x;
  if (i < n) p[i] = 0.0f;
}

extern "C" void kernel_launch(void* const* d_in, const int* in_sizes, int n_in,
                              void* d_out, int out_size, void* d_ws, size_t ws_size,
                              hipStream_t stream) {
  const float* x        = (const float*)d_in[0];
  const float* fh_W1    = (const float*)d_in[1];
  const float* fh_b1    = (const float*)d_in[2];
  const float* fh_bn1   = (const float*)d_in[3];
  const float* fh_W2    = (const float*)d_in[4];
  const float* fh_b2    = (const float*)d_in[5];
  const float* fh_bn2   = (const float*)d_in[6];
  const float* conv_W1  = (const float*)d_in[7];   // [4,64,64]
  const float* conv_b1  = (const float*)d_in[8];   // [4,64]
  const float* conv_bn1 = (const float*)d_in[9];   // [4,4,64]
  const float* conv_W2  = (const float*)d_in[10];
  const float* conv_b2  = (const float*)d_in[11];
  const float* conv_bn2 = (const float*)d_in[12];
  const float* lin_W    = (const float*)d_in[13];  // [5,64,10]
  const float* lin_b    = (const float*)d_in[14];  // [5,10]
  const int*   eidx     = (const int*)d_in[15];    // [2, N_EDGES]
  const int*   batch    = (const int*)d_in[16];
  float* out = (float*)d_out;

  const int* src = eidx;
  const int* dst = eidx + N_EDGES;

  float* ws     = (float*)d_ws;
  float* h      = ws;                              // 6,400,000 f
  float* agg    = ws + (size_t)N_NODES * 64;       // 6,400,000 f
  float* pooled = agg + (size_t)N_NODES * 64;      // 5*1000*64 = 320,000 f
  float* counts = pooled + 5 * N_GRAPHS * 64;      // 1000 f

  const int accN = 5 * N_GRAPHS * 64 + N_GRAPHS;
  gin_zero_kernel<<<(accN + 255) / 256, 256, 0, stream>>>(pooled, accN);

  const int mlpBlocks = (NTILES + 3) / 4;          // 4 waves (tiles) per 128-thr block

  // layer 0: encoder MLP, then pool (with node counts for the per-node bias)
  gin_mlp_kernel<D_FEAT><<<mlpBlocks, 128, 0, stream>>>(
      x, fh_W1, fh_b1, fh_bn1, fh_W2, fh_b2, fh_bn2, h, NTILES);
  gin_pool_kernel<<<(N_NODES * 16 + 255) / 256, 256, 0, stream>>>(
      h, batch, pooled, counts, 1);

  // layers 1..4: agg = h + segment_sum(h[src] -> dst); h = MLP(agg); pool
  for (int l = 0; l < 4; l++) {
    hipMemcpyAsync(agg, h, (size_t)N_NODES * 64 * sizeof(float),
                   hipMemcpyDeviceToDevice, stream);
    gin_scatter_kernel<<<(int)(((long long)N_EDGES * 16 + 255) / 256), 256, 0, stream>>>(
        h, src, dst, agg);
    gin_mlp_kernel<HID><<<mlpBlocks, 128, 0, stream>>>(
        agg,
        conv_W1 + (size_t)l * 64 * 64, conv_b1 + l * 64, conv_bn1 + l * 256,
        conv_W2 + (size_t)l * 64 * 64, conv_b2 + l * 64, conv_bn2 + l * 256,
        h, NTILES);
    gin_pool_kernel<<<(N_NODES * 16 + 255) / 256, 256, 0, stream>>>(
        h, batch, pooled + (size_t)(l + 1) * N_GRAPHS * 64, counts, 0);
  }

  gin_final_kernel<<<(N_GRAPHS * 10 + 255) / 256, 256, 0, stream>>>(
      pooled, counts, lin_W, lin_b, out);
}